// AdaptiveTokenMerger_60120952209556
// MI455X (gfx1250) — compile-verified
//
#include <hip/hip_runtime.h>
#include <hip/hip_bf16.h>

typedef __attribute__((ext_vector_type(2))) float v2f;
typedef __attribute__((ext_vector_type(8))) float v8f;

#define DEVINL __device__ __forceinline__

// V_WMMA_F32_16X16X4_F32: A=16x4 f32 (2 VGPR/lane), B=4x16 f32, C/D=16x16 f32 (8 VGPR/lane)
DEVINL v8f wmma_f32x4(v2f a, v2f b, v8f c) {
  return __builtin_amdgcn_wmma_f32_16x16x4_f32(false, a, false, b, (short)0, c, false, false);
}

DEVINL float apply_act(int ACT, float v) {
  if (ACT == 1) {                       // gelu, tanh approximation (jax default)
    float t = 0.7978845608028654f * (v + 0.044715f * v * v * v);
    v = 0.5f * v * (1.0f + tanhf(t));
  } else if (ACT == 2) {                // silu
    v = v / (1.0f + expf(-v));
  }
  return v;
}

// --------------------------- GEMM (WMMA f32) ---------------------------
// C[M,N] = A[M,K] @ op(B) + bias, op(B) = B[K,N] (TRANSB=0) or B[N,K]^T (TRANSB=1)
// ACT: 0=none, 1=gelu(tanh), 2=silu.  GUARD: 0 = tile always in-bounds (branchless),
// 1 = clamp+select edge handling (branchless too, no exec-mask churn in the k-loop).
// Slice-batched over grid.z: gs = sliceStart+z, bb=gs/Hdiv, hh=gs%Hdiv; aLocal/cLocal
// address chunk-local score scratch by z instead.
template<int ACT, int GUARD>
DEVINL void store_tile(float* __restrict__ C, const float* __restrict__ bias,
                       v8f acc, int row0, int col0, int M, int N, int ldc, int lane) {
  int c = col0 + (lane & 15);
  if (GUARD && c >= N) return;
  float bv = bias ? bias[c] : 0.0f;
  int rbase = row0 + ((lane >> 4) << 3);
#pragma unroll
  for (int r = 0; r < 8; ++r) {
    int rr = rbase + r;
    if (!GUARD || rr < M) {
      C[(long)rr * ldc + c] = apply_act(ACT, acc[r] + bv);
    }
  }
}

template<int TRANSB, int ACT, int GUARD>
__global__ __launch_bounds__(128)
void gemm_wmma_kernel(const float* __restrict__ A, const float* __restrict__ B,
                      float* __restrict__ C, const float* __restrict__ bias,
                      int M, int N, int K, int lda, int ldb, int ldc,
                      int Hdiv, int sliceStart,
                      long sAb, long sAh, long sBb, long sBh, long sCb, long sCh,
                      int aLocal, int cLocal) {
  int z = blockIdx.z;
  long gs = (long)sliceStart + z;
  long bb = gs / Hdiv, hh = gs % Hdiv;
  const float* Ap = A + (aLocal ? (long)z * sAb : bb * sAb + hh * sAh);
  const float* Bp = B + bb * sBb + hh * sBh;
  float*       Cp = C + (cLocal ? (long)z * sCb : bb * sCb + hh * sCh);

  int lane = threadIdx.x & 31;
  int wid  = threadIdx.x >> 5;               // 4 waves: 2x2 arrangement, 32x32 per wave
  int row0 = blockIdx.y * 64 + (wid >> 1) * 32;
  int col0 = blockIdx.x * 64 + (wid & 1) * 32;

  v8f acc00 = {0.f,0.f,0.f,0.f,0.f,0.f,0.f,0.f};
  v8f acc01 = acc00, acc10 = acc00, acc11 = acc00;
  const v2f zero2 = {0.f, 0.f};

  int aRow = lane & 15;                       // M within tile
  int kOff = (lane >> 4) << 1;                // lanes 16-31 hold K+2,K+3
  int cSel = lane & 15;                       // N within tile

  // Loop-invariant validity + clamped coordinates (GUARD path is branch-free:
  // loads use clamped in-bounds addresses, invalid frags are zeroed by select).
  int r0 = row0 + aRow,      r1 = row0 + 16 + aRow;
  int c0 = col0 + cSel,      c1 = col0 + 16 + cSel;
  bool vr0 = true, vr1 = true, vc0 = true, vc1 = true;
  if (GUARD) {
    vr0 = r0 < M; vr1 = r1 < M; vc0 = c0 < N; vc1 = c1 < N;
    int mm = M - 1, nn = N - 1;
    r0 = vr0 ? r0 : mm;  r1 = vr1 ? r1 : mm;
    c0 = vc0 ? c0 : nn;  c1 = vc1 ? c1 : nn;
  }

  const float* pa0 = Ap + (long)r0 * lda + kOff;
  const float* pa1 = Ap + (long)r1 * lda + kOff;
  const float* pb0;
  const float* pb1;
  long bstep;
  if (TRANSB) {            // B[N,K]^T : contiguous along K
    pb0 = Bp + (long)c0 * ldb + kOff;
    pb1 = Bp + (long)c1 * ldb + kOff;
    bstep = 4;
  } else {                 // B[K,N]   : strided along K
    pb0 = Bp + (long)kOff * ldb + c0;
    pb1 = Bp + (long)kOff * ldb + c1;
    bstep = 4L * ldb;
  }

  for (int k0 = 0; k0 < K; k0 += 4) {         // K is always a multiple of 4 here
    v2f a0 = *(const v2f*)pa0;
    v2f a1 = *(const v2f*)pa1;
    v2f b0, b1;
    if (TRANSB) {
      b0 = *(const v2f*)pb0;
      b1 = *(const v2f*)pb1;
    } else {
      b0.x = pb0[0]; b0.y = pb0[ldb];
      b1.x = pb1[0]; b1.y = pb1[ldb];
    }
    if (GUARD) {
      a0 = vr0 ? a0 : zero2;  a1 = vr1 ? a1 : zero2;
      b0 = vc0 ? b0 : zero2;  b1 = vc1 ? b1 : zero2;
    }
    acc00 = wmma_f32x4(a0, b0, acc00);
    acc01 = wmma_f32x4(a0, b1, acc01);
    acc10 = wmma_f32x4(a1, b0, acc10);
    acc11 = wmma_f32x4(a1, b1, acc11);
    pa0 += 4; pa1 += 4; pb0 += bstep; pb1 += bstep;
  }
  store_tile<ACT, GUARD>(Cp, bias, acc00, row0,      col0,      M, N, ldc, lane);
  store_tile<ACT, GUARD>(Cp, bias, acc01, row0,      col0 + 16, M, N, ldc, lane);
  store_tile<ACT, GUARD>(Cp, bias, acc10, row0 + 16, col0,      M, N, ldc, lane);
  store_tile<ACT, GUARD>(Cp, bias, acc11, row0 + 16, col0 + 16, M, N, ldc, lane);
}

// --------------------------- LayerNorm ---------------------------
__global__ __launch_bounds__(256)
void layernorm_kernel(const float* __restrict__ x, const float* __restrict__ g,
                      const float* __restrict__ be, float* __restrict__ y, int C) {
  long row = blockIdx.x;
  const float* xr = x + row * (long)C;
  float* yr = y + row * (long)C;
  __shared__ float s1[256], s2[256];
  float a = 0.f, b = 0.f;
  for (int c = threadIdx.x; c < C; c += 256) { float v = xr[c]; a += v; b += v * v; }
  s1[threadIdx.x] = a; s2[threadIdx.x] = b; __syncthreads();
  for (int off = 128; off > 0; off >>= 1) {
    if ((int)threadIdx.x < off) { s1[threadIdx.x] += s1[threadIdx.x + off]; s2[threadIdx.x] += s2[threadIdx.x + off]; }
    __syncthreads();
  }
  float mean = s1[0] / C;
  float var  = s2[0] / C - mean * mean;
  float rstd = rsqrtf(var + 1e-5f);
  for (int c = threadIdx.x; c < C; c += 256) yr[c] = (xr[c] - mean) * rstd * g[c] + be[c];
}

// --------------------------- Row L2-normalize ---------------------------
__global__ __launch_bounds__(256)
void rownorm_kernel(const float* __restrict__ x, float* __restrict__ y, int C) {
  long row = blockIdx.x;
  const float* xr = x + row * (long)C;
  float* yr = y + row * (long)C;
  __shared__ float s1[256];
  float a = 0.f;
  for (int c = threadIdx.x; c < C; c += 256) { float v = xr[c]; a += v * v; }
  s1[threadIdx.x] = a; __syncthreads();
  for (int off = 128; off > 0; off >>= 1) {
    if ((int)threadIdx.x < off) s1[threadIdx.x] += s1[threadIdx.x + off];
    __syncthreads();
  }
  float rn = rsqrtf(s1[0]);
  for (int c = threadIdx.x; c < C; c += 256) yr[c] = xr[c] * rn;
}

// --------------------------- Row softmax of scale*x ---------------------------
__global__ __launch_bounds__(256)
void softmax_kernel(float* __restrict__ s, int cols, float scale) {
  long row = blockIdx.x;
  float* r = s + row * (long)cols;
  __shared__ float red[256];
  float m = -3.4e38f;
  for (int c = threadIdx.x; c < cols; c += 256) m = fmaxf(m, r[c]);
  red[threadIdx.x] = m; __syncthreads();
  for (int off = 128; off > 0; off >>= 1) {
    if ((int)threadIdx.x < off) red[threadIdx.x] = fmaxf(red[threadIdx.x], red[threadIdx.x + off]);
    __syncthreads();
  }
  m = red[0]; __syncthreads();
  float sum = 0.f;
  for (int c = threadIdx.x; c < cols; c += 256) {
    float e = expf(scale * (r[c] - m));
    r[c] = e; sum += e;
  }
  red[threadIdx.x] = sum; __syncthreads();
  for (int off = 128; off > 0; off >>= 1) {
    if ((int)threadIdx.x < off) red[threadIdx.x] += red[threadIdx.x + off];
    __syncthreads();
  }
  float inv = 1.f / red[0];
  for (int c = threadIdx.x; c < cols; c += 256) r[c] *= inv;
}

// --------------------------- Row max + argmax (first occurrence) ---------------------------
__global__ __launch_bounds__(256)
void rowargmax_kernel(const float* __restrict__ s, float* __restrict__ vmax,
                      int* __restrict__ imax, int cols) {
  long row = blockIdx.x;
  const float* r = s + row * (long)cols;
  __shared__ float sv[256]; __shared__ int si[256];
  float bm = -3.4e38f; int bi = 0x7fffffff;
  for (int c = threadIdx.x; c < cols; c += 256) {
    float v = r[c];
    if (v > bm) { bm = v; bi = c; }           // per-thread columns ascend -> first occurrence
  }
  sv[threadIdx.x] = bm; si[threadIdx.x] = bi; __syncthreads();
  for (int off = 128; off > 0; off >>= 1) {
    if ((int)threadIdx.x < off) {
      float ov = sv[threadIdx.x + off]; int oi = si[threadIdx.x + off];
      if (ov > sv[threadIdx.x] || (ov == sv[threadIdx.x] && oi < si[threadIdx.x])) {
        sv[threadIdx.x] = ov; si[threadIdx.x] = oi;
      }
    }
    __syncthreads();
  }
  if (threadIdx.x == 0) { vmax[row] = sv[0]; imax[row] = si[0]; }
}

// ---------- Stable descending argsort of 512 values (== jnp.argsort(-x)) ----------
__global__ __launch_bounds__(256)
void argsort_desc512_kernel(const float* __restrict__ nmax, int* __restrict__ edge) {
  int b = blockIdx.x;
  __shared__ float v[512]; __shared__ int id[512];
  for (int i = threadIdx.x; i < 512; i += 256) { v[i] = nmax[b * 512 + i]; id[i] = i; }
  __syncthreads();
  for (int ksz = 2; ksz <= 512; ksz <<= 1) {
    for (int j = ksz >> 1; j > 0; j >>= 1) {
      for (int t = threadIdx.x; t < 512; t += 256) {
        int ixj = t ^ j;
        if (ixj > t) {
          bool up = ((t & ksz) == 0);
          // desired final order: value desc, index asc on ties (stable)
          bool before = (v[t] > v[ixj]) || (v[t] == v[ixj] && id[t] < id[ixj]);
          if (up ? !before : before) {
            float tv = v[t]; v[t] = v[ixj]; v[ixj] = tv;
            int ti = id[t]; id[t] = id[ixj]; id[ixj] = ti;
          }
        }
      }
      __syncthreads();
    }
  }
  for (int i = threadIdx.x; i < 512; i += 256) edge[b * 512 + i] = id[i];
}

// --------------------------- Token merge ---------------------------
// out rows 0..495   : src rows 2*edge[b][16+t]   (unmerged, sorted order)
// out rows 496..1007: dst rows 2*(t-496)+1
__global__ __launch_bounds__(256)
void merge_gather_kernel(const float* __restrict__ x1, const int* __restrict__ edge,
                         float* __restrict__ xm, float* __restrict__ sm, int C) {
  long rowo = blockIdx.x;
  long b = rowo / 1008, t = rowo % 1008;
  int srcRow = (t < 496) ? 2 * edge[b * 512 + 16 + t] : 2 * (int)(t - 496) + 1;
  const float* src = x1 + (b * 1024 + srcRow) * (long)C;
  float* dst = xm + rowo * (long)C;
  for (int c = threadIdx.x; c < C; c += 256) dst[c] = src[c];
  if (threadIdx.x == 0) sm[rowo] = 1.0f;
}

__global__ __launch_bounds__(256)
void merge_scatter_kernel(const float* __restrict__ x1, const int* __restrict__ edge,
                          const int* __restrict__ nodeIdx, float* __restrict__ xm,
                          float* __restrict__ sm, int C) {
  int b = blockIdx.x >> 4, s = blockIdx.x & 15;   // 16 merged tokens per batch (R=16)
  int srcTok = edge[b * 512 + s];
  int dstTok = nodeIdx[b * 512 + srcTok];
  const float* src = x1 + ((long)b * 1024 + 2 * srcTok) * (long)C;
  float* dst = xm + ((long)b * 1008 + 496 + dstTok) * (long)C;
  for (int c = threadIdx.x; c < C; c += 256) atomicAdd(&dst[c], src[c]);
  if (threadIdx.x == 0) atomicAdd(&sm[(long)b * 1008 + 496 + dstTok], 1.0f);
}

__global__ __launch_bounds__(256)
void divrow_kernel(float* __restrict__ xm, const float* __restrict__ sm, int C) {
  long row = blockIdx.x;
  float inv = 1.f / sm[row];
  float* r = xm + row * (long)C;
  for (int c = threadIdx.x; c < C; c += 256) r[c] *= inv;
}

// --------------------------- Pair pooling ---------------------------
__global__ __launch_bounds__(256)
void poolhalf_kernel(const float* __restrict__ in, float* __restrict__ out, int Nout, int C) {
  long ob = blockIdx.x;
  long b = ob / Nout, i = ob % Nout;
  const float* p0 = in + (b * 2 * Nout + 2 * i) * (long)C;
  const float* p1 = p0 + C;
  float* o = out + ob * (long)C;
  for (int c = threadIdx.x; c < C; c += 256) o[c] = 0.5f * (p0[c] + p1[c]);
}

// --------------------------- o = a + beta*b ---------------------------
__global__ __launch_bounds__(256)
void axpy_kernel(const float* __restrict__ a, const float* __restrict__ b,
                 float* __restrict__ o, long n, float beta) {
  long i = (long)blockIdx.x * 256 + threadIdx.x;
  if (i < n) o[i] = a[i] + beta * b[i];
}

// --------------------------- Host-side launcher ---------------------------
struct GemmArgs {
  const float *A, *B, *bias; float* C;
  int M, N, K, lda, ldb, ldc;
  int slices, Hdiv, sliceStart;
  long sAb, sAh, sBb, sBh, sCb, sCh;
  int aLocal, cLocal;
};

static void run_gemm(hipStream_t st, int transb, int act, const GemmArgs& g) {
  dim3 grid((g.N + 63) / 64, (g.M + 63) / 64, g.slices), block(128);
  int guard = ((g.M & 63) != 0) || ((g.N & 63) != 0);
#define GCALL(TB, AC, GD) gemm_wmma_kernel<TB, AC, GD><<<grid, block, 0, st>>>( \
    g.A, g.B, g.C, g.bias, g.M, g.N, g.K, g.lda, g.ldb, g.ldc, g.Hdiv, g.sliceStart, \
    g.sAb, g.sAh, g.sBb, g.sBh, g.sCb, g.sCh, g.aLocal, g.cLocal)
#define GSEL(TB, AC) do { if (guard) GCALL(TB, AC, 1); else GCALL(TB, AC, 0); } while (0)
  if (transb == 0) { if (act == 0) GSEL(0, 0); else if (act == 1) GSEL(0, 1); else GSEL(0, 2); }
  else             { if (act == 0) GSEL(1, 0); else if (act == 1) GSEL(1, 1); else GSEL(1, 2); }
#undef GSEL
#undef GCALL
}

extern "C" void kernel_launch(void* const* d_in, const int* in_sizes, int n_in,
                              void* d_out, int out_size, void* d_ws, size_t ws_size,
                              hipStream_t stream) {
  const float* x    = (const float*)d_in[0];
  const float* g1   = (const float*)d_in[1];
  const float* be1  = (const float*)d_in[2];
  const float* Wqkv = (const float*)d_in[3];
  const float* bqkv = (const float*)d_in[4];
  const float* Wo   = (const float*)d_in[5];
  const float* bo   = (const float*)d_in[6];
  const float* g2   = (const float*)d_in[7];
  const float* be2  = (const float*)d_in[8];
  const float* W1   = (const float*)d_in[9];
  const float* bm1  = (const float*)d_in[10];
  const float* W2   = (const float*)d_in[11];
  const float* bm2  = (const float*)d_in[12];
  const float* Wp   = (const float*)d_in[13];
  const float* bp   = (const float*)d_in[14];
  const float* Wq   = (const float*)d_in[15];
  const float* bq   = (const float*)d_in[16];
  const float* Wk   = (const float*)d_in[17];
  const float* bk   = (const float*)d_in[18];
  const float* Wv   = (const float*)d_in[19];
  const float* bv   = (const float*)d_in[20];
  const float* Wmo  = (const float*)d_in[21];
  const float* bmo  = (const float*)d_in[22];
  const float* Wf1  = (const float*)d_in[23];
  const float* bf1  = (const float*)d_in[24];
  const float* Wf2  = (const float*)d_in[25];
  const float* bf2  = (const float*)d_in[26];
  float* out = (float*)d_out;

  // ---- workspace arena ----
  size_t off = 0;
  auto alloc = [&](size_t bytes) -> void* {
    void* p = (char*)d_ws + off;
    off += (bytes + 255) & ~(size_t)255;
    return p;
  };
  float* b_ln   = (float*)alloc(8192ull * 1024 * 4);   // LN output (reused LN1/LN2)
  float* b_big  = (float*)alloc(33030144ull * 4);      // qkv[8192x3072] / mlp-h[8064x4096] / f1[4032x4096]
  float* b_hc   = (float*)alloc(8192ull * 1024 * 4);   // head-concat / mnorm / mlp_out
  float* b_xat  = (float*)alloc(8192ull * 1024 * 4);   // x_attn -> xm/x2 -> k4
  float* b_x1   = (float*)alloc(8192ull * 1024 * 4);   // x1 -> x3
  float* b_msc  = (float*)alloc(8ull * 512 * 512 * 4); // merge cosine scores
  float* b_nmax = (float*)alloc(8ull * 512 * 4);
  int*   b_nidx = (int*)  alloc(8ull * 512 * 4);
  int*   b_edge = (int*)  alloc(8ull * 512 * 4);
  float* b_sm   = (float*)alloc(8ull * 1008 * 4);
  float* b_pool = (float*)alloc(4032ull * 1024 * 4);   // q_in -> kp
  float* b_q4   = (float*)alloc(4032ull * 1024 * 4);   // q4 -> combined
  float* b_mq   = (float*)alloc(4032ull * 1024 * 4);
  float* b_mk   = (float*)alloc(4032ull * 64 * 4);
  float* b_mv   = (float*)alloc(4032ull * 64 * 4);
  float* b_ctx  = (float*)alloc(4032ull * 1024 * 4);
  float* b_out5 = (float*)alloc(4032ull * 1024 * 4);
  float* b_att  = (float*)((char*)d_ws + off);          // attention-score chunks
  size_t remain = (ws_size > off) ? (ws_size - off) : 0;

  GemmArgs g;

  // 1) h = LN(x)                        [8192,1024]
  layernorm_kernel<<<8192, 256, 0, stream>>>(x, g1, be1, b_ln, 1024);

  // 2) qkv = h @ Wqkv + bqkv            [8192,3072]
  g = {}; g.A = b_ln; g.B = Wqkv; g.bias = bqkv; g.C = b_big;
  g.M = 8192; g.N = 3072; g.K = 1024; g.lda = 1024; g.ldb = 3072; g.ldc = 3072;
  g.slices = 1; g.Hdiv = 1;
  run_gemm(stream, 0, 0, g);

  // 3) MHA: per (b,h) scores -> softmax -> @v, chunked through scratch
  {
    size_t sliceB = 1024ull * 1024 * 4;
    int CH = (int)(remain / sliceB); if (CH < 1) CH = 1; if (CH > 32) CH = 32;
    for (int s0 = 0; s0 < 128; s0 += CH) {
      int ns = (128 - s0 < CH) ? (128 - s0) : CH;
      g = {}; g.A = b_big; g.B = b_big + 1024; g.C = b_att; g.bias = nullptr;
      g.M = 1024; g.N = 1024; g.K = 64; g.lda = 3072; g.ldb = 3072; g.ldc = 1024;
      g.slices = ns; g.Hdiv = 16; g.sliceStart = s0;
      g.sAb = 1024ll * 3072; g.sAh = 64; g.sBb = 1024ll * 3072; g.sBh = 64;
      g.sCb = 1024ll * 1024; g.cLocal = 1;
      run_gemm(stream, 1, 0, g);                               // q @ k^T
      softmax_kernel<<<ns * 1024, 256, 0, stream>>>(b_att, 1024, 0.125f);
      g = {}; g.A = b_att; g.B = b_big + 2048; g.C = b_hc; g.bias = nullptr;
      g.M = 1024; g.N = 64; g.K = 1024; g.lda = 1024; g.ldb = 3072; g.ldc = 1024;
      g.slices = ns; g.Hdiv = 16; g.sliceStart = s0;
      g.sAb = 1024ll * 1024; g.aLocal = 1;
      g.sBb = 1024ll * 3072; g.sBh = 64;
      g.sCb = 1024ll * 1024; g.sCh = 64;
      run_gemm(stream, 0, 0, g);                               // attn @ v -> head concat
    }
  }

  // 4) x_attn = hc @ Wo + bo            [8192,1024]
  g = {}; g.A = b_hc; g.B = Wo; g.bias = bo; g.C = b_xat;
  g.M = 8192; g.N = 1024; g.K = 1024; g.lda = 1024; g.ldb = 1024; g.ldc = 1024;
  g.slices = 1; g.Hdiv = 1;
  run_gemm(stream, 0, 0, g);

  // 5) x1 = x + x_attn
  axpy_kernel<<<(8388608 + 255) / 256, 256, 0, stream>>>(x, b_xat, b_x1, 8388608, 1.0f);

  // 6) mnorm = x_attn / ||x_attn||
  rownorm_kernel<<<8192, 256, 0, stream>>>(b_xat, b_hc, 1024);

  // 7) scores[b,i,j] = mnorm[b,2i] . mnorm[b,2j+1]   [8,512,512]
  g = {}; g.A = b_hc; g.B = b_hc + 1024; g.C = b_msc; g.bias = nullptr;
  g.M = 512; g.N = 512; g.K = 1024; g.lda = 2048; g.ldb = 2048; g.ldc = 512;
  g.slices = 8; g.Hdiv = 1;
  g.sAb = 1024ll * 1024; g.sBb = 1024ll * 1024; g.sCb = 512ll * 512;
  run_gemm(stream, 1, 0, g);

  // 8) node_max / node_idx ; 9) stable descending argsort per batch
  rowargmax_kernel<<<8 * 512, 256, 0, stream>>>(b_msc, b_nmax, b_nidx, 512);
  argsort_desc512_kernel<<<8, 256, 0, stream>>>(b_nmax, b_edge);

  // 10-12) merge -> xm (into b_xat), sm ; x2 = xm / sm
  merge_gather_kernel<<<8 * 1008, 256, 0, stream>>>(b_x1, b_edge, b_xat, b_sm, 1024);
  merge_scatter_kernel<<<8 * 16, 256, 0, stream>>>(b_x1, b_edge, b_nidx, b_xat, b_sm, 1024);
  divrow_kernel<<<8 * 1008, 256, 0, stream>>>(b_xat, b_sm, 1024);

  // 13) LN2 ; 14) gelu MLP up ; 15) MLP down ; 16) x3 = x2 + mlp
  layernorm_kernel<<<8064, 256, 0, stream>>>(b_xat, g2, be2, b_ln, 1024);
  g = {}; g.A = b_ln; g.B = W1; g.bias = bm1; g.C = b_big;
  g.M = 8064; g.N = 4096; g.K = 1024; g.lda = 1024; g.ldb = 4096; g.ldc = 4096;
  g.slices = 1; g.Hdiv = 1;
  run_gemm(stream, 0, 1, g);
  g = {}; g.A = b_big; g.B = W2; g.bias = bm2; g.C = b_hc;
  g.M = 8064; g.N = 1024; g.K = 4096; g.lda = 4096; g.ldb = 1024; g.ldc = 1024;
  g.slices = 1; g.Hdiv = 1;
  run_gemm(stream, 0, 0, g);
  axpy_kernel<<<(8257536 + 255) / 256, 256, 0, stream>>>(b_xat, b_hc, b_x1, 8257536, 1.0f);

  // 17) k4 = x3 @ Wp + bp               [8064,1024]
  g = {}; g.A = b_x1; g.B = Wp; g.bias = bp; g.C = b_xat;
  g.M = 8064; g.N = 1024; g.K = 1024; g.lda = 1024; g.ldb = 1024; g.ldc = 1024;
  g.slices = 1; g.Hdiv = 1;
  run_gemm(stream, 0, 0, g);

  // 18) q_in = pool(x3) ; 19) q4 = q_in @ Wp + bp ; 20) kp = pool(k4) ; 21) combined = q4 + 2*kp
  poolhalf_kernel<<<8 * 504, 256, 0, stream>>>(b_x1, b_pool, 504, 1024);
  g = {}; g.A = b_pool; g.B = Wp; g.bias = bp; g.C = b_q4;
  g.M = 4032; g.N = 1024; g.K = 1024; g.lda = 1024; g.ldb = 1024; g.ldc = 1024;
  g.slices = 1; g.Hdiv = 1;
  run_gemm(stream, 0, 0, g);
  poolhalf_kernel<<<8 * 504, 256, 0, stream>>>(b_xat, b_pool, 504, 1024);
  axpy_kernel<<<(4128768 + 255) / 256, 256, 0, stream>>>(b_q4, b_pool, b_q4, 4128768, 2.0f);

  // 22-24) mq / mk / mv
  g = {}; g.A = b_q4; g.B = Wq; g.bias = bq; g.C = b_mq;
  g.M = 4032; g.N = 1024; g.K = 1024; g.lda = 1024; g.ldb = 1024; g.ldc = 1024;
  g.slices = 1; g.Hdiv = 1;
  run_gemm(stream, 0, 0, g);
  g = {}; g.A = b_q4; g.B = Wk; g.bias = bk; g.C = b_mk;
  g.M = 4032; g.N = 64; g.K = 1024; g.lda = 1024; g.ldb = 64; g.ldc = 64;
  g.slices = 1; g.Hdiv = 1;
  run_gemm(stream, 0, 0, g);
  g = {}; g.A = b_q4; g.B = Wv; g.bias = bv; g.C = b_mv;
  g.M = 4032; g.N = 64; g.K = 1024; g.lda = 1024; g.ldb = 64; g.ldc = 64;
  g.slices = 1; g.Hdiv = 1;
  run_gemm(stream, 0, 0, g);

  // 25) cross attention per (b,h): mq @ mk^T -> softmax -> @ mv -> ctx
  {
    size_t sliceB = 504ull * 504 * 4;
    int CH = (int)(remain / sliceB); if (CH < 1) CH = 1; if (CH > 64) CH = 64;
    for (int s0 = 0; s0 < 128; s0 += CH) {
      int ns = (128 - s0 < CH) ? (128 - s0) : CH;
      g = {}; g.A = b_mq; g.B = b_mk; g.C = b_att; g.bias = nullptr;
      g.M = 504; g.N = 504; g.K = 64; g.lda = 1024; g.ldb = 64; g.ldc = 504;
      g.slices = ns; g.Hdiv = 16; g.sliceStart = s0;
      g.sAb = 504ll * 1024; g.sAh = 64; g.sBb = 504ll * 64;
      g.sCb = 504ll * 504; g.cLocal = 1;
      run_gemm(stream, 1, 0, g);
      softmax_kernel<<<ns * 504, 256, 0, stream>>>(b_att, 504, 0.125f);
      g = {}; g.A = b_att; g.B = b_mv; g.C = b_ctx; g.bias = nullptr;
      g.M = 504; g.N = 64; g.K = 504; g.lda = 504; g.ldb = 64; g.ldc = 1024;
      g.slices = ns; g.Hdiv = 16; g.sliceStart = s0;
      g.sAb = 504ll * 504; g.aLocal = 1;
      g.sBb = 504ll * 64;
      g.sCb = 504ll * 1024; g.sCh = 64;
      run_gemm(stream, 0, 0, g);
    }
  }

  // 26) out5 = ctx @ Wmo + bmo ; 27) f1 = silu(out5 @ Wf1 + bf1) ; 28) out = f1 @ Wf2 + bf2
  g = {}; g.A = b_ctx; g.B = Wmo; g.bias = bmo; g.C = b_out5;
  g.M = 4032; g.N = 1024; g.K = 1024; g.lda = 1024; g.ldb = 1024; g.ldc = 1024;
  g.slices = 1; g.Hdiv = 1;
  run_gemm(stream, 0, 0, g);
  g = {}; g.A = b_out5; g.B = Wf1; g.bias = bf1; g.C = b_big;
  g.M = 4032; g.N = 4096; g.K = 1024; g.lda = 1024; g.ldb = 4096; g.ldc = 4096;
  g.slices = 1; g.Hdiv = 1;
  run_gemm(stream, 0, 2, g);
  g = {}; g.A = b_big; g.B = Wf2; g.bias = bf2; g.C = out;
  g.M = 4032; g.N = 1024; g.K = 4096; g.lda = 4096; g.ldb = 1024; g.ldc = 1024;
  g.slices = 1; g.Hdiv = 1;
  run_gemm(stream, 0, 0, g);
}